// GCN_28028956574228
// MI455X (gfx1250) — compile-verified
//
#include <hip/hip_runtime.h>
#include <hip/hip_bf16.h>

// ---------------------------------------------------------------------------
// GCN (3-layer, improved self-loops) for MI455X / gfx1250.
//  - Dense per-layer GEMM x@W done with V_WMMA_F32_16X16X4_F32 (full fp32).
//  - Edge gather/scale/scatter via coalesced loads + global_atomic_add_f32
//    (the true bottleneck; working set ~50MB lives in the 192MB L2).
// ---------------------------------------------------------------------------

typedef __attribute__((ext_vector_type(2))) float v2f;
typedef __attribute__((ext_vector_type(8))) float v8f;

#define GCN_N 50000
#define GCN_E 1250000
#define GCN_D 64
#define GCN_L 3
#define IMPROVED_FILL 2.0f

// ---------------- graph-normalization kernels ----------------

__global__ void gcn_deg_init(float* __restrict__ deg) {
    int n = blockIdx.x * blockDim.x + threadIdx.x;
    if (n < GCN_N) deg[n] = IMPROVED_FILL;   // self-loop weight 2.0
}

__global__ void gcn_deg_scatter(const int* __restrict__ col,
                                const float* __restrict__ w,
                                float* __restrict__ deg) {
    int e = blockIdx.x * blockDim.x + threadIdx.x;
    if (e < GCN_E) atomicAdd(&deg[col[e]], w[e]);
}

// in-place deg -> deg^{-1/2}; also emit self-loop norm = 2 * dis^2
__global__ void gcn_dis(float* __restrict__ deg,
                        float* __restrict__ norm_self) {
    int n = blockIdx.x * blockDim.x + threadIdx.x;
    if (n < GCN_N) {
        float d = deg[n];
        float dis = (d > 0.0f) ? rsqrtf(d) : 0.0f;
        deg[n] = dis;
        norm_self[n] = IMPROVED_FILL * dis * dis;
    }
}

__global__ void gcn_edge_norm(const int* __restrict__ row,
                              const int* __restrict__ col,
                              const float* __restrict__ w,
                              const float* __restrict__ dis,
                              float* __restrict__ norm) {
    int e = blockIdx.x * blockDim.x + threadIdx.x;
    if (e < GCN_E) norm[e] = dis[row[e]] * w[e] * dis[col[e]];
}

// ---------------- WMMA f32 GEMM: H[N,64] = X[N,64] @ W[64,64] ----------------
// One wave per 16-row block; 4 f32 accumulators cover all 64 output columns.
// K loop: 16 steps of K=4 using V_WMMA_F32_16X16X4_F32. 50000 = 3125 * 16.

#define GEMM_WAVES 4   // waves per block (blockDim = 128)

__global__ void gcn_gemm_wmma(const float* __restrict__ X,
                              const float* __restrict__ W,
                              float* __restrict__ H) {
    const int lane = threadIdx.x & 31;
    const int wave = threadIdx.x >> 5;
    const int mblk = blockIdx.x * GEMM_WAVES + wave;   // uniform per wave
    if (mblk >= GCN_N / 16) return;

    const int m0   = mblk * 16;
    const int l15  = lane & 15;
    const int kOff = (lane >> 4) * 2;      // lanes 16-31 carry K+2/K+3

    v8f acc0 = {}, acc1 = {}, acc2 = {}, acc3 = {};

    const float* xr = X + (m0 + l15) * GCN_D;

    #pragma unroll
    for (int k0 = 0; k0 < GCN_D; k0 += 4) {
        // A fragment (16x4): lane l<16 holds A[M=l][k0+0], A[M=l][k0+1];
        // lane l>=16 holds A[M=l-16][k0+2], A[M=l-16][k0+3].
        v2f a;
        a.x = xr[k0 + kOff + 0];
        a.y = xr[k0 + kOff + 1];

        // B fragments (4x16) for the 4 column tiles; same K striping.
        const float* w0 = W + (k0 + kOff) * GCN_D + l15;
        const float* w1 = w0 + GCN_D;

        v2f b0; b0.x = w0[ 0]; b0.y = w1[ 0];
        v2f b1; b1.x = w0[16]; b1.y = w1[16];
        v2f b2; b2.x = w0[32]; b2.y = w1[32];
        v2f b3; b3.x = w0[48]; b3.y = w1[48];

        acc0 = __builtin_amdgcn_wmma_f32_16x16x4_f32(false, a, false, b0,
                                                     (short)0, acc0, false, false);
        acc1 = __builtin_amdgcn_wmma_f32_16x16x4_f32(false, a, false, b1,
                                                     (short)0, acc1, false, false);
        acc2 = __builtin_amdgcn_wmma_f32_16x16x4_f32(false, a, false, b2,
                                                     (short)0, acc2, false, false);
        acc3 = __builtin_amdgcn_wmma_f32_16x16x4_f32(false, a, false, b3,
                                                     (short)0, acc3, false, false);
    }

    // C/D layout: VGPR r -> M=r (lanes 0-15), M=r+8 (lanes 16-31); N = lane&15.
    const int mrow = m0 + (lane >> 4) * 8;
    float* hr = H + mrow * GCN_D + l15;
    #pragma unroll
    for (int r = 0; r < 8; ++r) {
        hr[(r * GCN_D) +  0] = acc0[r];
        hr[(r * GCN_D) + 16] = acc1[r];
        hr[(r * GCN_D) + 32] = acc2[r];
        hr[(r * GCN_D) + 48] = acc3[r];
    }
}

// ---------------- aggregation kernels ----------------

// agg[n][d] = h[n][d] * norm_self[n]   (self-loop term; also zero-free init)
__global__ void gcn_agg_init(const float* __restrict__ h,
                             const float* __restrict__ norm_self,
                             float* __restrict__ agg) {
    int t = blockIdx.x * blockDim.x + threadIdx.x;
    if (t < GCN_N * GCN_D) {
        int n = t >> 6;
        agg[t] = h[t] * norm_self[n];
    }
}

// one thread per (edge, channel): coalesced gather + f32 atomic scatter-add
__global__ void gcn_edge_scatter(const int* __restrict__ row,
                                 const int* __restrict__ col,
                                 const float* __restrict__ norm,
                                 const float* __restrict__ h,
                                 float* __restrict__ agg) {
    long long t = (long long)blockIdx.x * blockDim.x + threadIdx.x;
    if (t < (long long)GCN_E * GCN_D) {
        int e = (int)(t >> 6);
        int d = (int)(t & 63);
        float v = h[(long long)row[e] * GCN_D + d] * norm[e];
        atomicAdd(&agg[(long long)col[e] * GCN_D + d], v);
    }
}

// x_out[n][d] = relu(agg[n][d] + b[d])
__global__ void gcn_bias_relu(const float* __restrict__ agg,
                              const float* __restrict__ b,
                              float* __restrict__ x_out) {
    int t = blockIdx.x * blockDim.x + threadIdx.x;
    if (t < GCN_N * GCN_D) {
        int d = t & 63;
        x_out[t] = fmaxf(agg[t] + b[d], 0.0f);
    }
}

// ---------------- final projection: out[n] = x[n,:] . Wf + bf ----------------
// wave32: one wave per node, 2 channels per lane, shfl_xor tree reduction.
__global__ void gcn_final_dot(const float* __restrict__ x,
                              const float* __restrict__ Wf,
                              const float* __restrict__ bf,
                              float* __restrict__ out) {
    int lane = threadIdx.x & 31;
    int wave = threadIdx.x >> 5;
    int n = blockIdx.x * (blockDim.x >> 5) + wave;
    if (n >= GCN_N) return;
    const float* xr = x + (long long)n * GCN_D;
    float p = xr[lane] * Wf[lane] + xr[lane + 32] * Wf[lane + 32];
    #pragma unroll
    for (int s = 16; s >= 1; s >>= 1)
        p += __shfl_xor(p, s, 32);
    if (lane == 0) out[n] = p + bf[0];
}

// ---------------- launch orchestration ----------------

extern "C" void kernel_launch(void* const* d_in, const int* in_sizes, int n_in,
                              void* d_out, int out_size, void* d_ws, size_t ws_size,
                              hipStream_t stream) {
    const float* x0   = (const float*)d_in[0];              // [N, 64]
    const int*   eidx = (const int*)d_in[1];                // [2, E]
    const float* ew   = (const float*)d_in[2];              // [E]
    const float* Ws   = (const float*)d_in[3];              // [3, 64, 64]
    const float* bs   = (const float*)d_in[4];              // [3, 64]
    const float* Wf   = (const float*)d_in[5];              // [64, 1]
    const float* bf   = (const float*)d_in[6];              // [1]
    float* out = (float*)d_out;                             // [N, 1]

    const int* row = eidx;            // edge_index[0]
    const int* col = eidx + GCN_E;    // edge_index[1]

    // workspace layout
    char* ws = (char*)d_ws;
    float* deg       = (float*)ws;                 ws += (size_t)GCN_N * 4;          // then dis
    float* norm_self = (float*)ws;                 ws += (size_t)GCN_N * 4;
    float* norm      = (float*)ws;                 ws += (size_t)GCN_E * 4;
    float* h         = (float*)ws;                 ws += (size_t)GCN_N * GCN_D * 4;
    float* agg       = (float*)ws;                 ws += (size_t)GCN_N * GCN_D * 4;
    float* xbuf      = (float*)ws;                 ws += (size_t)GCN_N * GCN_D * 4;
    (void)ws_size; (void)in_sizes; (void)n_in; (void)out_size;

    const int B = 256;
    const int gN  = (GCN_N + B - 1) / B;
    const int gE  = (GCN_E + B - 1) / B;
    const int gND = (GCN_N * GCN_D + B - 1) / B;
    const long long totED = (long long)GCN_E * GCN_D;
    const int gED = (int)((totED + B - 1) / B);
    const int gGemm = (GCN_N / 16 + GEMM_WAVES - 1) / GEMM_WAVES;

    // normalization (graph-structure only; computed once)
    gcn_deg_init<<<gN, B, 0, stream>>>(deg);
    gcn_deg_scatter<<<gE, B, 0, stream>>>(col, ew, deg);
    gcn_dis<<<gN, B, 0, stream>>>(deg, norm_self);          // deg now holds dis
    gcn_edge_norm<<<gE, B, 0, stream>>>(row, col, ew, deg, norm);

    const float* xin = x0;
    for (int l = 0; l < GCN_L; ++l) {
        gcn_gemm_wmma<<<gGemm, 32 * GEMM_WAVES, 0, stream>>>(xin, Ws + l * GCN_D * GCN_D, h);
        gcn_agg_init<<<gND, B, 0, stream>>>(h, norm_self, agg);
        gcn_edge_scatter<<<gED, B, 0, stream>>>(row, col, norm, h, agg);
        gcn_bias_relu<<<gND, B, 0, stream>>>(agg, bs + l * GCN_D, xbuf);
        xin = xbuf;
    }

    gcn_final_dot<<<(GCN_N + 7) / 8, B, 0, stream>>>(xbuf, Wf, bf, out);
}